// AdaptiveComputationModule_62637803045452
// MI455X (gfx1250) — compile-verified
//
#include <hip/hip_runtime.h>
#include <math.h>

typedef __attribute__((ext_vector_type(2))) float v2f;
typedef __attribute__((ext_vector_type(4))) float v4f;
typedef __attribute__((ext_vector_type(8))) float v8f;

#define BROWS   8192
#define H       1024
#define HQ      256
#define TM      32        // rows per workgroup
#define STRIDE  1028      // LDS row pitch in floats (pad 4 -> conflict-free b64 reads)
#define NSTEPS  10
#define THRESH  0.95f

// Branch-free tanh: v_exp_f32 + v_rcp_f32, no EXEC divergence.
// tanh(x) = sign(x) * (1 - 2/(exp(2|x|)+1)); exp overflow -> rcp(inf)=0 -> 1. ~2 ulp.
__device__ __forceinline__ float fast_tanh(float x) {
    float ax = fabsf(x);
    float e  = __expf(2.0f * ax);                       // v_exp_f32 (fast)
    float t  = 1.0f - 2.0f * __builtin_amdgcn_rcpf(e + 1.0f);
    return copysignf(t, x);
}

__device__ __forceinline__ float fast_sigmoid(float x) {
    // 1/(1+exp(-x)); exp(-x) overflow -> rcp(inf)=0. ~2 ulp.
    return __builtin_amdgcn_rcpf(1.0f + __expf(-x));
}

// One-time weight repack: Wp[kg*N + n] = float4{ W[4kg+0][n], W[4kg+1][n], W[4kg+2][n], W[4kg+3][n] }
// This makes each B-fragment of v_wmma_f32_16x16x4_f32 a single aligned float2 load:
//   lanes 0-15 take (K, K+1), lanes 16-31 take (K+2, K+3)  [ISA 16x4 f32 A/B layout]
__global__ void pack4_kernel(const float* __restrict__ W, float* __restrict__ Wp,
                             int N, int total) {
    int i = blockIdx.x * blockDim.x + threadIdx.x;
    if (i >= total) return;
    int kg = i / N;
    int n  = i - kg * N;
    const float* src = W + (size_t)kg * 4 * N + n;
    v4f v;
    v.x = src[0];
    v.y = src[(size_t)N];
    v.z = src[(size_t)2 * N];
    v.w = src[(size_t)3 * N];
    *(v4f*)(Wp + (size_t)i * 4) = v;
}

__launch_bounds__(256, 1)
__global__ void act_kernel(const float* __restrict__ x,
                           const float* __restrict__ Wcp,   // packed Wc
                           const float* __restrict__ bc,
                           const float* __restrict__ W1p,   // packed W1
                           const float* __restrict__ b1,
                           const float* __restrict__ W2,
                           const float* __restrict__ b2,
                           float* __restrict__ out) {
    __shared__ float xcbuf[2][TM * STRIDE];    // double-buffered recurrent state (f32)
    __shared__ float pscr[8 * 32 * 8];         // per-(wave,lane,vgpr) halting partials
    __shared__ float s_cum[TM], s_rem[TM], s_run[TM], s_uw[TM];

    const int tid  = threadIdx.x;
    const int wid  = tid >> 5;
    const int lane = tid & 31;
    const int ln   = lane & 15;
    const int hi   = lane >> 4;                 // which half of the wave
    const int row0 = blockIdx.x * TM;

    if (tid < TM) { s_cum[tid] = 0.f; s_rem[tid] = 0.f; s_run[tid] = 1.f; }

    // Load x tile into LDS buffer 0 (float4, rows padded to STRIDE)
    for (int i = tid; i < TM * (H / 4); i += 256) {
        int r  = i >> 8;         // H/4 == 256
        int c4 = i & 255;
        v4f v = *(const v4f*)(x + (size_t)(row0 + r) * H + (size_t)c4 * 4);
        *(v4f*)&xcbuf[0][r * STRIDE + c4 * 4] = v;
    }
    __syncthreads();

    // Wave tiling: 2 row-groups (16 rows) x 4 col-groups
    const int rg   = wid & 1;
    const int cg   = wid >> 1;
    const int arow = rg * 16 + ln;              // A-fragment row for this lane
    const int aofs = arow * STRIDE + 2 * hi;    // + (k0) added in loop

    int cur = 0;
    const v8f vzero = {0.f, 0.f, 0.f, 0.f, 0.f, 0.f, 0.f, 0.f};

    for (int step = 0; step < NSTEPS; ++step) {
        const float* __restrict__ cb = xcbuf[cur];
        float* __restrict__ nb       = xcbuf[cur ^ 1];

        // ---------- main GEMM: nb = tanh(cb @ Wc + bc), 4 chunks of 256 cols ----------
        for (int ch = 0; ch < 4; ++ch) {
            const int nbase = ch * 256 + cg * 64;
            v8f acc[4] = {vzero, vzero, vzero, vzero};
            for (int k0 = 0; k0 < H; k0 += 4) {
                v2f a = *(const v2f*)&cb[aofs + k0];
#pragma unroll
                for (int t = 0; t < 4; ++t) {
                    const int col = nbase + t * 16 + ln;
                    v2f b = *(const v2f*)(Wcp + ((size_t)(k0 >> 2) * H + col) * 4 + 2 * hi);
                    acc[t] = __builtin_amdgcn_wmma_f32_16x16x4_f32(
                        false, a, false, b, (short)0, acc[t], false, false);
                }
            }
#pragma unroll
            for (int t = 0; t < 4; ++t) {
                const int col = nbase + t * 16 + ln;
                const float bcv = bc[col];
#pragma unroll
                for (int r = 0; r < 8; ++r) {
                    // D layout: vgpr r, lanes 0-15 -> M=r, lanes 16-31 -> M=r+8
                    nb[(rg * 16 + r + 8 * hi) * STRIDE + col] = fast_tanh(acc[t][r] + bcv);
                }
            }
        }
        __syncthreads();

        // ---------- halting: h = relu(nb @ W1 + b1); partial p = h @ W2 ----------
        {
            v8f acc[4] = {vzero, vzero, vzero, vzero};
            for (int k0 = 0; k0 < H; k0 += 4) {
                v2f a = *(const v2f*)&nb[aofs + k0];
#pragma unroll
                for (int t = 0; t < 4; ++t) {
                    const int col = cg * 64 + t * 16 + ln;
                    v2f b = *(const v2f*)(W1p + ((size_t)(k0 >> 2) * HQ + col) * 4 + 2 * hi);
                    acc[t] = __builtin_amdgcn_wmma_f32_16x16x4_f32(
                        false, a, false, b, (short)0, acc[t], false, false);
                }
            }
            float part[8] = {0.f, 0.f, 0.f, 0.f, 0.f, 0.f, 0.f, 0.f};
#pragma unroll
            for (int t = 0; t < 4; ++t) {
                const int col = cg * 64 + t * 16 + ln;
                const float b1v = b1[col];
                const float w2v = W2[col];
#pragma unroll
                for (int r = 0; r < 8; ++r) {
                    float h = acc[t][r] + b1v;
                    h = h > 0.f ? h : 0.f;
                    part[r] += h * w2v;
                }
            }
#pragma unroll
            for (int r = 0; r < 8; ++r)
                pscr[(wid * 32 + lane) * 8 + r] = part[r];
        }
        __syncthreads();

        // ---------- deterministic p reduction + per-row ACT bookkeeping ----------
        if (tid < TM) {
            const int row = tid;
            const int rrg = row >> 4;
            const int q   = row & 15;
            const int h2  = q >> 3;
            const int rr  = q & 7;
            float sum = 0.f;
#pragma unroll
            for (int c = 0; c < 4; ++c) {
                const int w = rrg | (c << 1);            // wave with rg=rrg, cg=c
                const int base = (w * 32 + h2 * 16) * 8 + rr;
#pragma unroll
                for (int l = 0; l < 16; ++l) sum += pscr[base + l * 8];
            }
            const float p = fast_sigmoid(sum + b2[0]);
            float cum = s_cum[row], rem = s_rem[row], run = s_run[row];
            const float pm = p * run;
            const float nc = cum + pm;
            const bool halt = (nc >= THRESH) && (run > 0.5f);
            if (halt) rem = 1.f - cum;
            if (run > 0.5f) cum = fminf(nc, 1.f);
            const float uw = halt ? rem : pm;
            if (halt) run = 0.f;
            s_cum[row] = cum; s_rem[row] = rem; s_run[row] = run; s_uw[row] = uw;
        }
        __syncthreads();

        // ---------- out += uw * xc_new (exclusive tile; step 0 initializes) ----------
        for (int i = tid; i < TM * (H / 4); i += 256) {
            int r  = i >> 8;
            int c4 = i & 255;
            v4f xv = *(const v4f*)&nb[r * STRIDE + c4 * 4];
            const float uw = s_uw[r];
            float* op = out + (size_t)(row0 + r) * H + (size_t)c4 * 4;
            if (step == 0) {
                v4f o = xv * uw;
                *(v4f*)op = o;
            } else {
                v4f o = *(const v4f*)op;
                o += xv * uw;
                *(v4f*)op = o;
            }
        }
        cur ^= 1;
        __syncthreads();
    }

    // ---------- never-halted remainder: out += (1-cum)*running * xc_final ----------
    if (tid < TM) s_uw[tid] = (1.f - s_cum[tid]) * s_run[tid];
    __syncthreads();
    const float* __restrict__ fb = xcbuf[cur];
    for (int i = tid; i < TM * (H / 4); i += 256) {
        int r  = i >> 8;
        int c4 = i & 255;
        v4f xv = *(const v4f*)&fb[r * STRIDE + c4 * 4];
        const float uw = s_uw[r];
        float* op = out + (size_t)(row0 + r) * H + (size_t)c4 * 4;
        v4f o = *(const v4f*)op;
        o += xv * uw;
        *(v4f*)op = o;
    }
}

extern "C" void kernel_launch(void* const* d_in, const int* in_sizes, int n_in,
                              void* d_out, int out_size, void* d_ws, size_t ws_size,
                              hipStream_t stream) {
    const float* x  = (const float*)d_in[0];
    const float* Wc = (const float*)d_in[1];
    const float* bc = (const float*)d_in[2];
    const float* W1 = (const float*)d_in[3];
    const float* b1 = (const float*)d_in[4];
    const float* W2 = (const float*)d_in[5];
    const float* b2 = (const float*)d_in[6];
    float* out = (float*)d_out;

    float* Wcp = (float*)d_ws;                       // 1024*1024 floats (4 MB)
    float* W1p = Wcp + (size_t)H * H;                // 1024*256 floats (1 MB)

    const int totalC = (H / 4) * H;                  // 262144
    pack4_kernel<<<(totalC + 255) / 256, 256, 0, stream>>>(Wc, Wcp, H, totalC);
    const int total1 = (H / 4) * HQ;                 // 65536
    pack4_kernel<<<(total1 + 255) / 256, 256, 0, stream>>>(W1, W1p, HQ, total1);

    act_kernel<<<BROWS / TM, 256, 0, stream>>>(x, Wcp, bc, W1p, b1, W2, b2, out);
}